// RelationalMessagePassingModule_45973329937221
// MI455X (gfx1250) — compile-verified
//
#include <hip/hip_runtime.h>
#include <hip/hip_bf16.h>
#include <math.h>

// ---------------------------------------------------------------------------
// CDNA5 (gfx1250) implementation of the relational message-passing network.
//  - fp32 WMMA (v_wmma_f32_16x16x4_f32) for all MLP GEMMs
//  - weights pre-transposed once per call -> B-frags load as aligned b64
//  - gather staged in LDS, tile indices cached in LDS for the scatter phase
//  - smooth-max scatter via monotone-key atomicMax(u32) + atomicAdd(f32)
//  - messages recomputed in the exp pass (no 205MB message buffer)
//  - rational mish: x * (p^2-1)/(p^2+1), p = 1+e^x  (one v_exp, no tanhf)
// ---------------------------------------------------------------------------

typedef float v2f __attribute__((ext_vector_type(2)));
typedef float v8f __attribute__((ext_vector_type(8)));

#define SMOOTH 12.0f
#define KEY_NEG_INF 0x007FFFFFu   // key(-inf) under the monotone mapping

__device__ __forceinline__ v8f wmma4(v2f a, v2f b, v8f c) {
    // D(16x16,f32) = A(16x4,f32) * B(4x16,f32) + C
    return __builtin_amdgcn_wmma_f32_16x16x4_f32(
        /*neg_a=*/false, a, /*neg_b=*/false, b,
        /*c_mod=*/(short)0, c, /*reuse_a=*/false, /*reuse_b=*/false);
}

// mish(x) = x * tanh(softplus(x)) = x * (p^2-1)/(p^2+1), p = 1+e^x
__device__ __forceinline__ float mish(float x) {
    if (x > 20.0f) return x;           // tanh(softplus(x)) == 1 in fp32
    float p  = 1.0f + __expf(x);
    float p2 = p * p;
    return x * __fdividef(p2 - 1.0f, p2 + 1.0f);
}

// monotone float -> u32 key (order-preserving, so atomicMax(u32) == float max)
__device__ __forceinline__ unsigned f2key(float f) {
    unsigned u = __float_as_uint(f);
    return (u & 0x80000000u) ? ~u : (u | 0x80000000u);
}

// ---------------------------------------------------------------------------
// init: node_emb = 0, maxkey = key(-inf), sum = 1e-16
// ---------------------------------------------------------------------------
__global__ void init_kernel(float* __restrict__ emb,
                            unsigned* __restrict__ maxkey,
                            float* __restrict__ sumbuf, int n) {
    int i = blockIdx.x * blockDim.x + threadIdx.x;
    if (i < n) {
        emb[i] = 0.0f;
        maxkey[i] = KEY_NEG_INF;
        sumbuf[i] = 1e-16f;
    }
}

// out[c*R + r] = in[r*C + c]  (R x C -> C x R), run once per call per matrix
__global__ void transpose_kernel(const float* __restrict__ in,
                                 float* __restrict__ out, int R, int C) {
    int i = blockIdx.x * blockDim.x + threadIdx.x;
    if (i < R * C) {
        int r = i / C, c = i - r * C;
        out[c * R + r] = in[i];
    }
}

// decode max keys in place: key -> float, key(-inf) -> 0.0 (include_self=False)
__global__ void decode_kernel(unsigned* __restrict__ buf, int n) {
    int i = blockIdx.x * blockDim.x + threadIdx.x;
    if (i < n) {
        unsigned k = buf[i];
        float f;
        if (k == KEY_NEG_INF) {
            f = 0.0f;
        } else {
            unsigned u = (k & 0x80000000u) ? (k & 0x7FFFFFFFu) : ~k;
            f = __uint_as_float(u);
        }
        buf[i] = __float_as_uint(f);
    }
}

// ---------------------------------------------------------------------------
// Relation pass: one wave owns 16 atoms.
//   X (16 x D) = gathered node embeddings   (LDS)
//   H = mish(X @ Wi + bi)                   (LDS)
//   MSG = X + H @ Wo + bo
//   SUMPASS=false: scatter atomicMax(key(MSG))
//   SUMPASS=true : scatter atomicAdd(exp(12*(MSG - max[idx])))
// WiT/WoT are transposed: WT[n*D + k].
// ---------------------------------------------------------------------------
template <int D, int WPB, bool SUMPASS>
__launch_bounds__(WPB * 32)
__global__ void rel_pass_kernel(const float* __restrict__ emb,
                                const int* __restrict__ idx,
                                const float* __restrict__ WiT,
                                const float* __restrict__ bi,
                                const float* __restrict__ WoT,
                                const float* __restrict__ bo,
                                unsigned* __restrict__ maxkey,
                                const float* __restrict__ maxv,
                                float* __restrict__ sumbuf,
                                int num_atoms) {
    constexpr int ARITY = D / 64;
    __shared__ float Xs[WPB][16 * D];
    __shared__ float Hs[WPB][16 * D];
    __shared__ int   Is[WPB][16 * ARITY];

    const int wave = threadIdx.x >> 5;
    const int lane = threadIdx.x & 31;
    const int half = lane >> 4;   // 0: K=k0..k0+1, 1: K=k0+2..k0+3
    const int lid  = lane & 15;
    const int tile = blockIdx.x * WPB + wave;
    const int base = tile * 16;
    const bool active = base < num_atoms;

    float* X = Xs[wave];
    float* H = Hs[wave];
    int*   I = Is[wave];

    __builtin_prefetch(WiT, 0, 1);
    __builtin_prefetch(WoT, 0, 1);

    // ---- gather X rows (float4 per thread); cache tile indices in LDS -----
    if (active) {
        if (lane < 16 * ARITY) I[lane] = idx[base * ARITY + lane];
        constexpr int R4 = D / 4;  // float4s per row
        for (int t = lane; t < 16 * R4; t += 32) {
            int row = t / R4;
            int c4  = (t - row * R4) * 4;
            int node = idx[(base + row) * ARITY + (c4 >> 6)];
            float4 v = *reinterpret_cast<const float4*>(emb + node * 64 + (c4 & 63));
            *reinterpret_cast<float4*>(X + row * D + c4) = v;
        }
    }
    __syncthreads();

    // ---- H = mish(X @ Wi + bi) --------------------------------------------
    if (active) {
        for (int j = 0; j < D / 16; ++j) {
            v8f acc = {};
            const float* wrow = WiT + (size_t)(j * 16 + lid) * D;
            #pragma unroll 4
            for (int k0 = 0; k0 < D; k0 += 4) {
                int ka = k0 + 2 * half;
                v2f a = *reinterpret_cast<const v2f*>(X + lid * D + ka);
                v2f b = *reinterpret_cast<const v2f*>(wrow + ka);
                acc = wmma4(a, b, acc);
            }
            float bias = bi[j * 16 + lid];
            #pragma unroll
            for (int v = 0; v < 8; ++v)
                H[(v + 8 * half) * D + j * 16 + lid] = mish(acc[v] + bias);
        }
    }
    __syncthreads();

    // ---- MSG = X + H @ Wo + bo, scatter -----------------------------------
    if (active) {
        for (int j = 0; j < D / 16; ++j) {
            v8f acc = {};
            const float* wrow = WoT + (size_t)(j * 16 + lid) * D;
            #pragma unroll 4
            for (int k0 = 0; k0 < D; k0 += 4) {
                int ka = k0 + 2 * half;
                v2f a = *reinterpret_cast<const v2f*>(H + lid * D + ka);
                v2f b = *reinterpret_cast<const v2f*>(wrow + ka);
                acc = wmma4(a, b, acc);
            }
            int n_g  = j * 16 + lid;      // column within a*E
            int sub  = n_g >> 6;          // which tuple slot
            int c    = n_g & 63;          // feature within E
            float bias = bo[n_g];
            #pragma unroll
            for (int v = 0; v < 8; ++v) {
                int m = v + 8 * half;     // row within tile
                float msg = X[m * D + n_g] + acc[v] + bias;
                int node = I[m * ARITY + sub];
                int addr = node * 64 + c;
                if (SUMPASS) {
                    atomicAdd(sumbuf + addr,
                              __expf(SMOOTH * (msg - maxv[addr])));
                } else {
                    atomicMax(maxkey + addr, f2key(msg));
                }
            }
        }
    }
}

// ---------------------------------------------------------------------------
// Node update: U = [log(sum)/12 + max | emb] (16 x 128)
//              H = mish(U @ Wi + bi) (16 x 128)
//              emb += H @ Wo + bo    (16 x 64)
// Also resets maxkey/sum for the next layer. WiT/WoT transposed (stride 128).
// ---------------------------------------------------------------------------
template <int WPB>
__launch_bounds__(WPB * 32)
__global__ void update_kernel(float* __restrict__ emb,
                              const float* __restrict__ WiT,
                              const float* __restrict__ bi,
                              const float* __restrict__ WoT,
                              const float* __restrict__ bo,
                              unsigned* __restrict__ maxkey,  // decoded floats
                              float* __restrict__ sumbuf,
                              int num_nodes) {
    __shared__ float Us[WPB][16 * 128];
    __shared__ float Hs[WPB][16 * 128];
    const float* maxv = reinterpret_cast<const float*>(maxkey);

    const int wave = threadIdx.x >> 5;
    const int lane = threadIdx.x & 31;
    const int half = lane >> 4;
    const int lid  = lane & 15;
    const int tile = blockIdx.x * WPB + wave;
    const int base = tile * 16;
    const bool active = base < num_nodes;

    float* U = Us[wave];
    float* H = Hs[wave];

    // ---- build U ----------------------------------------------------------
    if (active) {
        for (int t = lane; t < 16 * 32; t += 32) {
            int row = t >> 5;
            int c4  = (t & 31) * 4;
            int node = base + row;
            float4 v;
            if (c4 < 64) {
                float4 s = *reinterpret_cast<const float4*>(sumbuf + node * 64 + c4);
                float4 m = *reinterpret_cast<const float4*>(maxv + node * 64 + c4);
                v.x = __logf(s.x) * (1.0f / SMOOTH) + m.x;
                v.y = __logf(s.y) * (1.0f / SMOOTH) + m.y;
                v.z = __logf(s.z) * (1.0f / SMOOTH) + m.z;
                v.w = __logf(s.w) * (1.0f / SMOOTH) + m.w;
            } else {
                v = *reinterpret_cast<const float4*>(emb + node * 64 + (c4 - 64));
            }
            *reinterpret_cast<float4*>(U + row * 128 + c4) = v;
        }
    }
    __syncthreads();

    // ---- H = mish(U @ Wi + bi), 128x128 -----------------------------------
    if (active) {
        for (int j = 0; j < 8; ++j) {
            v8f acc = {};
            const float* wrow = WiT + (size_t)(j * 16 + lid) * 128;
            #pragma unroll 4
            for (int k0 = 0; k0 < 128; k0 += 4) {
                int ka = k0 + 2 * half;
                v2f a = *reinterpret_cast<const v2f*>(U + lid * 128 + ka);
                v2f b = *reinterpret_cast<const v2f*>(wrow + ka);
                acc = wmma4(a, b, acc);
            }
            float bias = bi[j * 16 + lid];
            #pragma unroll
            for (int v = 0; v < 8; ++v)
                H[(v + 8 * half) * 128 + j * 16 + lid] = mish(acc[v] + bias);
        }
    }
    __syncthreads();

    // ---- emb += H @ Wo + bo, 128x64; reset accumulators -------------------
    if (active) {
        for (int j = 0; j < 4; ++j) {
            v8f acc = {};
            const float* wrow = WoT + (size_t)(j * 16 + lid) * 128;
            #pragma unroll 4
            for (int k0 = 0; k0 < 128; k0 += 4) {
                int ka = k0 + 2 * half;
                v2f a = *reinterpret_cast<const v2f*>(H + lid * 128 + ka);
                v2f b = *reinterpret_cast<const v2f*>(wrow + ka);
                acc = wmma4(a, b, acc);
            }
            int n_g = j * 16 + lid;
            float bias = bo[n_g];
            #pragma unroll
            for (int v = 0; v < 8; ++v) {
                int m = v + 8 * half;
                // residual: right half of U holds the old embedding
                emb[(base + m) * 64 + n_g] = U[m * 128 + 64 + n_g] + acc[v] + bias;
            }
        }
        // reset smooth-max state for the next layer
        for (int t = lane; t < 16 * 16; t += 32) {
            int row = t >> 4;
            int c4  = (t & 15) * 4;
            int node = base + row;
            uint4 ki = make_uint4(KEY_NEG_INF, KEY_NEG_INF, KEY_NEG_INF, KEY_NEG_INF);
            float4 si = make_float4(1e-16f, 1e-16f, 1e-16f, 1e-16f);
            *reinterpret_cast<uint4*>(maxkey + node * 64 + c4) = ki;
            *reinterpret_cast<float4*>(sumbuf + node * 64 + c4) = si;
        }
    }
}

// ---------------------------------------------------------------------------
// Host side
// ---------------------------------------------------------------------------
extern "C" void kernel_launch(void* const* d_in, const int* in_sizes, int n_in,
                              void* d_out, int out_size, void* d_ws, size_t ws_size,
                              hipStream_t stream) {
    // jax pytree flattening (sorted dict keys):
    // params.rel.{r1,r2a,r2b,r3}.{Wi,Wo,bi,bo}, params.upd.{Wi,Wo,bi,bo},
    // then r1_idx, r2a_idx, r2b_idx, r3_idx, node_sizes
    const float *Wi_[5], *Wo_[5], *bi_[5], *bo_[5];
    int p = 0;
    for (int r = 0; r < 5; ++r) {
        Wi_[r] = (const float*)d_in[p++];
        Wo_[r] = (const float*)d_in[p++];
        bi_[r] = (const float*)d_in[p++];
        bo_[r] = (const float*)d_in[p++];
    }
    const int* idx_[4];
    for (int r = 0; r < 4; ++r) idx_[r] = (const int*)d_in[p++];

    const int arity[4] = {1, 2, 2, 3};
    int natoms[4];
    for (int r = 0; r < 4; ++r) natoms[r] = in_sizes[20 + r] / arity[r];

    const int N  = out_size / 64;   // 50000 nodes
    const int NE = N * 64;

    float*    emb    = (float*)d_out;                          // node embeddings
    unsigned* maxkey = (unsigned*)d_ws;                        // NE u32
    float*    sumbuf = (float*)((char*)d_ws + (size_t)NE * 4); // NE f32
    float*    wt     = (float*)((char*)d_ws + 2 * (size_t)NE * 4);
    (void)ws_size; (void)n_in;

    // transpose all weight matrices into workspace (once per call)
    const int din_[5]  = {64, 128, 128, 192, 128};
    const int dout_[5] = {64, 128, 128, 192, 64};
    float *WiT_[5], *WoT_[5];
    {
        float* cur = wt;
        for (int r = 0; r < 5; ++r) {
            int nI = din_[r] * din_[r];
            int nO = din_[r] * dout_[r];
            WiT_[r] = cur; cur += nI;
            WoT_[r] = cur; cur += nO;
            transpose_kernel<<<(nI + 255) / 256, 256, 0, stream>>>(Wi_[r], WiT_[r], din_[r], din_[r]);
            transpose_kernel<<<(nO + 255) / 256, 256, 0, stream>>>(Wo_[r], WoT_[r], din_[r], dout_[r]);
        }
    }

    init_kernel<<<(NE + 255) / 256, 256, 0, stream>>>(emb, maxkey, sumbuf, NE);

    const float* maxv = (const float*)maxkey;

    for (int layer = 0; layer < 3; ++layer) {
        // ---- pass 1: scatter max of messages ----
        {
            int t0 = (natoms[0] + 15) / 16;
            rel_pass_kernel<64, 4, false><<<(t0 + 3) / 4, 128, 0, stream>>>(
                emb, idx_[0], WiT_[0], bi_[0], WoT_[0], bo_[0], maxkey, maxv, sumbuf, natoms[0]);
            int t1 = (natoms[1] + 15) / 16;
            rel_pass_kernel<128, 2, false><<<(t1 + 1) / 2, 64, 0, stream>>>(
                emb, idx_[1], WiT_[1], bi_[1], WoT_[1], bo_[1], maxkey, maxv, sumbuf, natoms[1]);
            int t2 = (natoms[2] + 15) / 16;
            rel_pass_kernel<128, 2, false><<<(t2 + 1) / 2, 64, 0, stream>>>(
                emb, idx_[2], WiT_[2], bi_[2], WoT_[2], bo_[2], maxkey, maxv, sumbuf, natoms[2]);
            int t3 = (natoms[3] + 15) / 16;
            rel_pass_kernel<192, 2, false><<<(t3 + 1) / 2, 64, 0, stream>>>(
                emb, idx_[3], WiT_[3], bi_[3], WoT_[3], bo_[3], maxkey, maxv, sumbuf, natoms[3]);
        }
        // ---- decode keys to floats (neginf -> 0) ----
        decode_kernel<<<(NE + 255) / 256, 256, 0, stream>>>(maxkey, NE);
        // ---- pass 2: recompute messages, scatter exp-sum ----
        {
            int t0 = (natoms[0] + 15) / 16;
            rel_pass_kernel<64, 4, true><<<(t0 + 3) / 4, 128, 0, stream>>>(
                emb, idx_[0], WiT_[0], bi_[0], WoT_[0], bo_[0], maxkey, maxv, sumbuf, natoms[0]);
            int t1 = (natoms[1] + 15) / 16;
            rel_pass_kernel<128, 2, true><<<(t1 + 1) / 2, 64, 0, stream>>>(
                emb, idx_[1], WiT_[1], bi_[1], WoT_[1], bo_[1], maxkey, maxv, sumbuf, natoms[1]);
            int t2 = (natoms[2] + 15) / 16;
            rel_pass_kernel<128, 2, true><<<(t2 + 1) / 2, 64, 0, stream>>>(
                emb, idx_[2], WiT_[2], bi_[2], WoT_[2], bo_[2], maxkey, maxv, sumbuf, natoms[2]);
            int t3 = (natoms[3] + 15) / 16;
            rel_pass_kernel<192, 2, true><<<(t3 + 1) / 2, 64, 0, stream>>>(
                emb, idx_[3], WiT_[3], bi_[3], WoT_[3], bo_[3], maxkey, maxv, sumbuf, natoms[3]);
        }
        // ---- node update MLP + reset accumulators ----
        {
            int tn = (N + 15) / 16;
            update_kernel<2><<<(tn + 1) / 2, 64, 0, stream>>>(
                emb, WiT_[4], bi_[4], WoT_[4], bo_[4], maxkey, sumbuf, N);
        }
    }
}